// GeometricEmbedding_23029614641193
// MI455X (gfx1250) — compile-verified
//
#include <hip/hip_runtime.h>

// ---- constants matching the reference ----
#define NRBF 32
#define RCUT 5.0f
static constexpr float INV_W   = 32.0f / 5.0f;          // 1/width, width = R_CUT/N_RBF
static constexpr float CSTEP   = 5.0f / 31.0f;          // linspace(0, R_CUT, 32) step
static constexpr float C1      = 0.48860251190291992f;  // sqrt(3/(4pi))
static constexpr float C2A     = 1.09254843059207907f;  // 0.5*sqrt(15/pi)
static constexpr float C2B     = 0.31539156525252005f;  // 0.25*sqrt(5/pi)
static constexpr float C2C     = 0.54627421529603953f;  // 0.25*sqrt(15/pi)
static constexpr float PI_F    = 3.14159265358979323846f;
static constexpr float INV_LAM = 1.0f;                  // 1/LAMBDA

typedef float v2f __attribute__((ext_vector_type(2)));
typedef float v8f __attribute__((ext_vector_type(8)));

#define BLOCK 256
#define WAVES (BLOCK / 32)

// ---------------------------------------------------------------------------
// Main pair kernel: one wave32 handles 32 consecutive pairs.
//  Phase 1: per-lane geometry (distance, cutoff, spherical harmonics).
//  Phase 2: lane-transposed RBF writes -> contiguous 512B per b128 store.
//  Phase 3: chi segment-sum on the matrix pipe (V_WMMA_F32_16X16X4_F32),
//           tails only do atomics.
// Requires n_pairs handled here to be a multiple of BLOCK (tail kernel below).
// ---------------------------------------------------------------------------
__global__ __launch_bounds__(BLOCK) void geom_pair_kernel(
    const float* __restrict__ R,
    const float* __restrict__ pair_mask,
    const int*   __restrict__ idx_i,
    const int*   __restrict__ idx_j,
    float*       __restrict__ rbf_out,
    float*       __restrict__ chi_out)
{
    __shared__ float xpose[WAVES][8][34];   // [wave][channel][pair], padded rows
    __shared__ int   seg_id[WAVES][16];

    const int tid  = threadIdx.x;
    const int lane = tid & 31;
    const int w    = tid >> 5;
    const int p    = blockIdx.x * BLOCK + tid;

    const int   i = idx_i[p];
    const int   j = idx_j[p];
    const float m = pair_mask[p];

    // gfx1250 prefetch of upcoming index/mask stream (global_prefetch_b8)
    __builtin_prefetch(idx_i + p + 8 * BLOCK, 0, 0);
    __builtin_prefetch(idx_j + p + 8 * BLOCK, 0, 0);
    __builtin_prefetch(pair_mask + p + 8 * BLOCK, 0, 0);

    const float rx = (R[3 * j + 0] - R[3 * i + 0]) * m;
    const float ry = (R[3 * j + 1] - R[3 * i + 1]) * m;
    const float rz = (R[3 * j + 2] - R[3 * i + 2]) * m;
    const float sq = rx * rx + ry * ry + rz * rz;
    float d = (sq == 0.0f) ? 0.0f : sqrtf(sq);
    d *= m;

    // ---- Phase 2: RBF expansion, transposed for coalescing ----
    // lane group g = lane>>3 handles pair (4t+g); lane&7 picks 4 centers.
    const int pbase = blockIdx.x * BLOCK + w * 32;
    const int c0    = (lane & 7) * 4;
    #pragma unroll
    for (int t = 0; t < 8; ++t) {
        const int   psub = t * 4 + (lane >> 3);
        const float dj   = __shfl(d, psub);
        const float mj   = __shfl(m, psub);
        float4 v;
        {
            float a0 = (dj - (float)(c0 + 0) * CSTEP) * INV_W;
            float a1 = (dj - (float)(c0 + 1) * CSTEP) * INV_W;
            float a2 = (dj - (float)(c0 + 2) * CSTEP) * INV_W;
            float a3 = (dj - (float)(c0 + 3) * CSTEP) * INV_W;
            v.x = expf(-0.5f * a0 * a0) * mj;
            v.y = expf(-0.5f * a1 * a1) * mj;
            v.z = expf(-0.5f * a2 * a2) * mj;
            v.w = expf(-0.5f * a3 * a3) * mj;
        }
        *(float4*)(&rbf_out[(size_t)(pbase + psub) * NRBF + c0]) = v;
    }

    // ---- Phase 3: weighted spherical harmonics + segment-sum ----
    const float inv_d = (d != 0.0f) ? (1.0f / d) : 0.0f;
    const float ux = rx * inv_d, uy = ry * inv_d, uz = rz * inv_d;
    float phi = 0.5f * (cosf(PI_F * d * (1.0f / RCUT)) + 1.0f);
    phi = (d < RCUT) ? phi : 0.0f;
    phi *= m;   // phi carries pair_mask -> weighted is exactly 0 for masked pairs

    float wv[8];
    wv[0] = C1 * uy;            wv[1] = C1 * uz;            wv[2] = C1 * ux;
    wv[3] = C2A * ux * uy;      wv[4] = C2A * uy * uz;
    wv[5] = C2B * (3.0f * uz * uz - 1.0f);
    wv[6] = C2A * ux * uz;      wv[7] = C2C * (ux * ux - uy * uy);
    #pragma unroll
    for (int c = 0; c < 8; ++c) wv[c] *= phi;

    // segment heads over sorted idx_i within the wave
    const int      prev  = __shfl_up(i, 1);
    const bool     head  = (lane == 0) || (prev != i);
    const unsigned hmask = (unsigned)__ballot(head);
    const int      nseg  = __popc(hmask);
    const unsigned lemsk = 0xffffffffu >> (31 - lane);
    const int      lseg  = __popc(hmask & lemsk) - 1;   // local segment id

    if (nseg <= 16) {           // wave-uniform branch -> EXEC all-ones for WMMA
        if (head) seg_id[w][lseg] = i;
        #pragma unroll
        for (int c = 0; c < 8; ++c) xpose[w][c][lane] = wv[c];
        __builtin_amdgcn_wave_barrier();   // keep DS stores before DS loads

        // D[seg][chan] = sum_p onehot[seg][p] * W[p][chan], K tiled by 4.
        // A 16x4 f32 layout: lanes0-15 M=lane K={0,1}; lanes16-31 M=lane-16 K={2,3}
        v8f acc = {};
        const int koff = (lane < 16) ? 0 : 2;
        const int srow = lane & 15;        // A: segment row / B,D: column
        #pragma unroll
        for (int t = 0; t < 8; ++t) {
            const int s0 = __shfl(lseg, 4 * t + koff);
            const int s1 = __shfl(lseg, 4 * t + koff + 1);
            v2f a;
            a[0] = (s0 == srow) ? 1.0f : 0.0f;
            a[1] = (s1 == srow) ? 1.0f : 0.0f;
            v2f b;
            if (srow < 8) {
                const float* src = &xpose[w][srow][4 * t + koff];
                b[0] = src[0];
                b[1] = src[1];
            } else {
                b[0] = 0.0f;
                b[1] = 0.0f;
            }
            acc = __builtin_amdgcn_wmma_f32_16x16x4_f32(
                false, a, false, b, (short)0, acc, false, false);
        }

        // D layout: lane c<16 -> column N=c rows 0..7; lane c+16 -> rows 8..15
        const int c = lane & 15;
        if (c < 8) {
            const int sbase = (lane < 16) ? 0 : 8;
            #pragma unroll
            for (int r = 0; r < 8; ++r) {
                const int s = sbase + r;
                if (s < nseg) {
                    atomicAdd(&chi_out[(size_t)seg_id[w][s] * 8 + c], acc[r]);
                }
            }
        }
    } else {
        // rare: >16 distinct points in one wave of sorted pairs
        #pragma unroll
        for (int c = 0; c < 8; ++c)
            atomicAdd(&chi_out[(size_t)i * 8 + c], wv[c]);
    }
}

// ---------------------------------------------------------------------------
// Scalar tail kernel for n_pairs % BLOCK leftovers (0 with harness sizes).
// ---------------------------------------------------------------------------
__global__ void geom_pair_tail_kernel(
    const float* __restrict__ R, const float* __restrict__ pair_mask,
    const int* __restrict__ idx_i, const int* __restrict__ idx_j,
    float* __restrict__ rbf_out, float* __restrict__ chi_out,
    int start, int n_pairs)
{
    const int p = start + blockIdx.x * 64 + threadIdx.x;
    if (p >= n_pairs) return;
    const int i = idx_i[p], j = idx_j[p];
    const float m = pair_mask[p];
    const float rx = (R[3 * j + 0] - R[3 * i + 0]) * m;
    const float ry = (R[3 * j + 1] - R[3 * i + 1]) * m;
    const float rz = (R[3 * j + 2] - R[3 * i + 2]) * m;
    const float sq = rx * rx + ry * ry + rz * rz;
    float d = (sq == 0.0f) ? 0.0f : sqrtf(sq);
    d *= m;
    for (int k = 0; k < NRBF; ++k) {
        float a = (d - (float)k * CSTEP) * INV_W;
        rbf_out[(size_t)p * NRBF + k] = expf(-0.5f * a * a) * m;
    }
    const float inv_d = (d != 0.0f) ? (1.0f / d) : 0.0f;
    const float ux = rx * inv_d, uy = ry * inv_d, uz = rz * inv_d;
    float phi = 0.5f * (cosf(PI_F * d * (1.0f / RCUT)) + 1.0f);
    phi = (d < RCUT) ? phi : 0.0f;
    phi *= m;
    float wv[8];
    wv[0] = C1 * uy;       wv[1] = C1 * uz;       wv[2] = C1 * ux;
    wv[3] = C2A * ux * uy; wv[4] = C2A * uy * uz;
    wv[5] = C2B * (3.0f * uz * uz - 1.0f);
    wv[6] = C2A * ux * uz; wv[7] = C2C * (ux * ux - uy * uy);
    for (int c = 0; c < 8; ++c)
        atomicAdd(&chi_out[(size_t)i * 8 + c], wv[c] * phi);
}

// ---------------------------------------------------------------------------
// chi finalize: multiply by point_mask / LAMBDA
// ---------------------------------------------------------------------------
__global__ void chi_scale_kernel(float* __restrict__ chi,
                                 const float* __restrict__ point_mask,
                                 int total)
{
    const int t = blockIdx.x * 256 + threadIdx.x;
    if (t < total) chi[t] *= point_mask[t >> 3] * INV_LAM;
}

extern "C" void kernel_launch(void* const* d_in, const int* in_sizes, int n_in,
                              void* d_out, int out_size, void* d_ws, size_t ws_size,
                              hipStream_t stream)
{
    const float* R          = (const float*)d_in[0];
    const float* pair_mask  = (const float*)d_in[1];
    const float* point_mask = (const float*)d_in[2];
    const int*   idx_i      = (const int*)d_in[3];
    const int*   idx_j      = (const int*)d_in[4];
    // d_in[5] = z, unused by the reference outputs

    const int n_pairs  = in_sizes[1];
    const int n_points = in_sizes[2];

    float* rbf_out = (float*)d_out;
    float* chi_out = rbf_out + (size_t)n_pairs * NRBF;

    // chi is accumulated with atomics -> must start at zero every launch
    hipMemsetAsync(chi_out, 0, (size_t)n_points * 8 * sizeof(float), stream);

    const int n_full = (n_pairs / BLOCK) * BLOCK;
    if (n_full > 0) {
        geom_pair_kernel<<<n_full / BLOCK, BLOCK, 0, stream>>>(
            R, pair_mask, idx_i, idx_j, rbf_out, chi_out);
    }
    const int rem = n_pairs - n_full;
    if (rem > 0) {
        geom_pair_tail_kernel<<<(rem + 63) / 64, 64, 0, stream>>>(
            R, pair_mask, idx_i, idx_j, rbf_out, chi_out, n_full, n_pairs);
    }
    const int total_chi = n_points * 8;
    chi_scale_kernel<<<(total_chi + 255) / 256, 256, 0, stream>>>(
        chi_out, point_mask, total_chi);
}